// BicliqueAttentionLayer_17197049053759
// MI455X (gfx1250) — compile-verified
//
#include <hip/hip_runtime.h>
#include <math.h>

typedef __attribute__((ext_vector_type(2))) float v2f;
typedef __attribute__((ext_vector_type(8))) float v8f;

#define IN_DIM 128
#define OUT_DIM 128

// ---------------------------------------------------------------------------
// Native CDNA5 no-return f32 atomic add at device scope (L2 atomic units).
// Fire-and-forget: tracked on STOREcnt; s_endpgm's implicit wait-idle plus
// stream ordering guarantees visibility to the next kernel.
// ---------------------------------------------------------------------------
__device__ __forceinline__ void atomic_add_f32_dev(float* p, float v) {
    asm volatile("global_atomic_add_f32 %0, %1, off scope:SCOPE_DEV"
                 :: "v"(p), "v"(v) : "memory");
}

// ---------------------------------------------------------------------------
// init: zero the output accumulator, set segmax=-inf, denom=0
// ---------------------------------------------------------------------------
__global__ void init_kernel(float* __restrict__ out, float* __restrict__ segmax,
                            float* __restrict__ denom, int nOut, int N) {
    int i = blockIdx.x * blockDim.x + threadIdx.x;
    int stride = gridDim.x * blockDim.x;
    for (int j = i; j < nOut; j += stride) out[j] = 0.0f;
    for (int j = i; j < N; j += stride) {
        segmax[j] = -INFINITY;
        denom[j]  = 0.0f;
    }
}

// ---------------------------------------------------------------------------
// h = (feat * mask) @ W^T   via V_WMMA_F32_16X16X4_F32 (exact fp32 math)
// One wave per 16x16 output tile: block = 256 threads = 8 waves = one
// 16-row x 128-col slab. K loop: 32 steps of K=4.
// ---------------------------------------------------------------------------
__global__ __launch_bounds__(256) void gemm_wmma_kernel(
    const float* __restrict__ feat, const float* __restrict__ mask,
    const float* __restrict__ W, float* __restrict__ h, int N) {
    const int wave = threadIdx.x >> 5;          // 0..7  -> n-tile
    const int lane = threadIdx.x & 31;
    const int m0   = blockIdx.x * 16;
    const int n0   = wave * 16;

    int row = m0 + (lane & 15);
    if (row > N - 1) row = N - 1;               // tail safety (reads only)
    const int n     = n0 + (lane & 15);
    const int khalf = (lane >= 16) ? 2 : 0;     // K split across lane halves

    const float* frow = feat + (size_t)row * IN_DIM;
    const float* wrow = W    + (size_t)n   * IN_DIM;

    v8f c = {};
    #pragma unroll 4
    for (int k = 0; k < IN_DIM; k += 4) {
        const int ko = k + khalf;
        v2f a  = *(const v2f*)(frow + ko);
        v2f b  = *(const v2f*)(wrow + ko);
        v2f mk = *(const v2f*)(mask + ko);
        a.x *= mk.x;
        a.y *= mk.y;
        c = __builtin_amdgcn_wmma_f32_16x16x4_f32(
                /*neg_a=*/false, a, /*neg_b=*/false, b,
                /*c_mod=*/(short)0, c, /*reuse_a=*/false, /*reuse_b=*/false);
    }

    // C/D layout: VGPR v, lane l -> row = v + (l>=16 ? 8 : 0), col = l&15
    const int col   = lane & 15;
    const int rbase = (lane >= 16) ? 8 : 0;
    float* hb = h + (size_t)(m0 + rbase) * OUT_DIM + n0 + col;
    if (m0 + 16 <= N) {                 // uniform branch: full tile, no exec churn
        #pragma unroll
        for (int v = 0; v < 8; ++v) hb[(size_t)v * OUT_DIM] = c[v];
    } else {                            // last partial tile only
        #pragma unroll
        for (int v = 0; v < 8; ++v)
            if (m0 + rbase + v < N) hb[(size_t)v * OUT_DIM] = c[v];
    }
}

// ---------------------------------------------------------------------------
// s = leaky_relu(h @ attn)   one wave per row, float4 per lane + wave reduce
// ---------------------------------------------------------------------------
__global__ __launch_bounds__(256) void score_kernel(
    const float* __restrict__ h, const float* __restrict__ attn,
    float* __restrict__ s, int N) {
    const int row  = (blockIdx.x << 3) + (threadIdx.x >> 5);
    const int lane = threadIdx.x & 31;
    if (row >= N) return;
    const float4 hv = ((const float4*)(h + (size_t)row * OUT_DIM))[lane];
    const float4 av = ((const float4*)attn)[lane];
    float d = hv.x * av.x + hv.y * av.y + hv.z * av.z + hv.w * av.w;
    #pragma unroll
    for (int off = 16; off > 0; off >>= 1) d += __shfl_down(d, off, 32);
    if (lane == 0) s[row] = (d > 0.0f) ? d : d * 0.01f;
}

// ---------------------------------------------------------------------------
// segment max of score over dst (ordered-int atomic trick, exact)
// ---------------------------------------------------------------------------
__global__ void segmax_kernel(const float* __restrict__ s,
                              const int* __restrict__ src,
                              const int* __restrict__ dst,
                              float* __restrict__ segmax, int E) {
    const int e = blockIdx.x * blockDim.x + threadIdx.x;
    if (e >= E) return;
    const float v = s[src[e]];
    int* a = (int*)(segmax + dst[e]);
    if (v >= 0.0f) atomicMax(a, __float_as_int(v));
    else           atomicMin((unsigned int*)a, (unsigned int)__float_as_int(v));
}

// ---------------------------------------------------------------------------
// edge accumulate: ex = exp(s[src]-segmax[dst]); denom[dst]+=ex;
// out[dst,:] += ex * h[src,:]    (one wave per edge, float4 per lane)
// h and out are both L2-resident (51 MB each vs 192 MB L2), so the gather
// and the 205M native f32 atomics all run at L2 speed.
// ---------------------------------------------------------------------------
__global__ __launch_bounds__(256) void edge_accum_kernel(
    const float* __restrict__ s, const float* __restrict__ h,
    const int* __restrict__ src, const int* __restrict__ dst,
    const float* __restrict__ segmax, float* __restrict__ denom,
    float* __restrict__ out, int E) {
    const int e    = (blockIdx.x << 3) + (threadIdx.x >> 5);
    const int lane = threadIdx.x & 31;
    if (e >= E) return;
    const int sv = src[e];
    const int dv = dst[e];
    const float ex = __expf(s[sv] - segmax[dv]);
    if (lane == 0) atomic_add_f32_dev(denom + dv, ex);
    const float4 hv = ((const float4*)(h + (size_t)sv * OUT_DIM))[lane];
    float* ob = out + (size_t)dv * OUT_DIM + lane * 4;
    atomic_add_f32_dev(ob + 0, ex * hv.x);
    atomic_add_f32_dev(ob + 1, ex * hv.y);
    atomic_add_f32_dev(ob + 2, ex * hv.z);
    atomic_add_f32_dev(ob + 3, ex * hv.w);
}

// ---------------------------------------------------------------------------
// finalize: out = relu(out / denom); empty segments (denom==0) -> 0
// ---------------------------------------------------------------------------
__global__ void finalize_kernel(float* __restrict__ out,
                                const float* __restrict__ denom, int N) {
    const int i = blockIdx.x * blockDim.x + threadIdx.x;   // one float4 each
    if (i >= N * (OUT_DIM / 4)) return;
    const int row = i >> 5;
    const float d = denom[row];
    const float inv = (d > 0.0f) ? 1.0f / d : 0.0f;
    float4* p = (float4*)out + i;
    float4 v = *p;
    v.x = fmaxf(v.x * inv, 0.0f);
    v.y = fmaxf(v.y * inv, 0.0f);
    v.z = fmaxf(v.z * inv, 0.0f);
    v.w = fmaxf(v.w * inv, 0.0f);
    *p = v;
}

// ---------------------------------------------------------------------------
extern "C" void kernel_launch(void* const* d_in, const int* in_sizes, int n_in,
                              void* d_out, int out_size, void* d_ws, size_t ws_size,
                              hipStream_t stream) {
    const float* feat = (const float*)d_in[0];
    const float* mask = (const float*)d_in[1];
    const float* W    = (const float*)d_in[2];
    const float* attn = (const float*)d_in[3];
    const int*   src  = (const int*)d_in[4];
    const int*   dst  = (const int*)d_in[5];

    const int N = in_sizes[0] / IN_DIM;   // 100000
    const int E = in_sizes[4];            // 1600000

    float* out    = (float*)d_out;        // doubles as the h_new accumulator
    float* h      = (float*)d_ws;         // N*128
    float* s      = h + (size_t)N * OUT_DIM;
    float* segmax = s + N;
    float* denom  = segmax + N;

    init_kernel<<<2048, 256, 0, stream>>>(out, segmax, denom, N * OUT_DIM, N);

    gemm_wmma_kernel<<<(N + 15) / 16, 256, 0, stream>>>(feat, mask, W, h, N);

    score_kernel<<<(N * 32 + 255) / 256, 256, 0, stream>>>(h, attn, s, N);

    segmax_kernel<<<(E + 255) / 256, 256, 0, stream>>>(s, src, dst, segmax, E);

    edge_accum_kernel<<<(E + 7) / 8, 256, 0, stream>>>(s, h, src, dst, segmax,
                                                       denom, out, E);

    finalize_kernel<<<(N * 32 + 255) / 256, 256, 0, stream>>>(out, denom, N);
}